// CustomLoss_60885456388844
// MI455X (gfx1250) — compile-verified
//
#include <hip/hip_runtime.h>
#include <math.h>

// ---------------------------------------------------------------------------
// CustomLoss on MI455X (gfx1250): memory-bound streaming reduction.
// 256 MB of fp32 logits -> HBM-bound (~11 us @ 23.3 TB/s). One wave32 per row.
// All 8 row-chunks are issued as async global->LDS transfers up-front
// (8 outstanding per wave = max MLP), drained with a descending
// s_wait_asynccnt ladder (async loads complete in order). Wave32 shuffle
// reductions, deterministic two-stage final reduction.
// ---------------------------------------------------------------------------

#define VOCAB 1024
#define ROWS_PER_BLOCK 8           // 8 waves * 32 lanes = 256 threads
#define CHUNK 128                  // floats per stage (32 lanes * float4)
#define NCHUNK (VOCAB / CHUNK)     // 8
#define EPSF 1e-10f

#if defined(__has_builtin)
#if __has_builtin(__builtin_amdgcn_global_load_async_to_lds_b128)
#define HAVE_ASYNC_LDS 1
#else
#define HAVE_ASYNC_LDS 0
#endif
#else
#define HAVE_ASYNC_LDS 0
#endif

// Pointer types matching the async-LDS builtin signature:
//   (v4i addrspace(1)*, v4i addrspace(3)*, Imm offset, Imm cpol)
typedef int v4i __attribute__((ext_vector_type(4)));
typedef __attribute__((address_space(1))) v4i* g_v4i_p;
typedef __attribute__((address_space(3))) v4i* l_v4i_p;

template <int N>
__device__ __forceinline__ void wait_asynccnt() {
#if __has_builtin(__builtin_amdgcn_s_wait_asynccnt)
  __builtin_amdgcn_s_wait_asynccnt(N);
#else
  asm volatile("s_wait_asynccnt %0" ::"i"(N) : "memory");
#endif
}

__device__ __forceinline__ void stage_async(const float* gsrc, float* ldst) {
#if HAVE_ASYNC_LDS
  __builtin_amdgcn_global_load_async_to_lds_b128(
      (g_v4i_p)gsrc, (l_v4i_p)ldst, /*offset=*/0, /*cpol=*/0);
#else
  (void)gsrc;
  (void)ldst;
#endif
}

// --- Kernel A: upper_bound = max(range_end - range_start) over vocab --------
__global__ void upper_bound_kernel(const int* __restrict__ range_start,
                                   const int* __restrict__ range_end,
                                   float* __restrict__ upper_out, int v) {
  __shared__ int sm[256];
  int t = threadIdx.x;
  int m = 0;
  for (int i = t; i < v; i += 256) m = max(m, range_end[i] - range_start[i]);
  sm[t] = m;
  __syncthreads();
  for (int s = 128; s > 0; s >>= 1) {
    if (t < s) sm[t] = max(sm[t], sm[t + s]);
    __syncthreads();
  }
  if (t == 0) *upper_out = (float)sm[0];
}

// --- Kernel B: per-row loss contribution, one wave32 per row ----------------
__global__ void loss_main_kernel(const float* __restrict__ logits,
                                 const int* __restrict__ gts,
                                 const int* __restrict__ event_ids,
                                 const int* __restrict__ range_start,
                                 const int* __restrict__ range_end,
                                 const float* __restrict__ upper_p,
                                 float* __restrict__ partials, int n_rows) {
  // Full row landing zone per wave: 8 waves * 1024 floats = 32 KB.
  __shared__ float stage_buf[ROWS_PER_BLOCK][NCHUNK][CHUNK];
  __shared__ float wave_out[ROWS_PER_BLOCK];

  const int lane = threadIdx.x & 31;
  const int wv = threadIdx.x >> 5;
  const int row = blockIdx.x * ROWS_PER_BLOCK + wv;

  float contrib = 0.0f;
  if (row < n_rows) {
    const int gt = gts[row];
    const int rs = range_start[gt];
    const int re = range_end[gt];
    const int evg = event_ids[gt];
    const float* __restrict__ rowp = logits + (size_t)row * VOCAB;

    float sum_all = 0.0f;  // sum exp over row
    float sum_in = 0.0f;   // sum exp over [rs, re)
    float vmax = -3.402823466e38f;
    int imax = 0;

    // Per-element accumulate (col ascending within a lane -> strict > keeps
    // the lowest column index on ties, matching jnp.argmax).
    auto accum = [&](int colbase, const float4& vv) {
      const float xs[4] = {vv.x, vv.y, vv.z, vv.w};
#pragma unroll
      for (int j = 0; j < 4; ++j) {
        const float x = xs[j];
        const int col = colbase + j;
        const float e = __expf(x);
        sum_all += e;
        if (col >= rs && col < re) sum_in += e;
        if (x > vmax) {
          vmax = x;
          imax = col;
        }
      }
    };

#if HAVE_ASYNC_LDS
    // Issue ALL chunk transfers up-front: 8 outstanding async loads per wave.
#pragma unroll
    for (int c = 0; c < NCHUNK; ++c)
      stage_async(rowp + c * CHUNK + lane * 4, &stage_buf[wv][c][lane * 4]);

    // Drain in order: async loads signal completion in issue order, so
    // ASYNCcnt <= (NCHUNK-1-c) implies chunks 0..c have landed in LDS.
#define CONSUME_CHUNK(c)                                                   \
  {                                                                        \
    wait_asynccnt<NCHUNK - 1 - (c)>();                                     \
    const float4 vv = *(const float4*)&stage_buf[wv][(c)][lane * 4];       \
    accum((c)*CHUNK + lane * 4, vv);                                       \
  }
    CONSUME_CHUNK(0)
    CONSUME_CHUNK(1)
    CONSUME_CHUNK(2)
    CONSUME_CHUNK(3)
    CONSUME_CHUNK(4)
    CONSUME_CHUNK(5)
    CONSUME_CHUNK(6)
    CONSUME_CHUNK(7)
#undef CONSUME_CHUNK
#else
#pragma unroll
    for (int c = 0; c < NCHUNK; ++c) {
      const float4 vv = *(const float4*)(rowp + c * CHUNK + lane * 4);
      accum(c * CHUNK + lane * 4, vv);
    }
#endif

    // Wave32 reductions: sums + argmax (ties -> lowest column index).
#pragma unroll
    for (int off = 16; off > 0; off >>= 1) {
      sum_all += __shfl_xor(sum_all, off, 32);
      sum_in += __shfl_xor(sum_in, off, 32);
      const float ov = __shfl_xor(vmax, off, 32);
      const int oi = __shfl_xor(imax, off, 32);
      if (ov > vmax || (ov == vmax && oi < imax)) {
        vmax = ov;
        imax = oi;
      }
    }

    if (lane == 0) {
      const int pred = imax;
      const bool same = (event_ids[pred] == evg);
      const float S = sum_all + EPSF;
      if (same) {
        // sum(mask1)=re-rs; mask2 term is exactly 0.
        const float cnt1 = (float)(re - rs);
        const float tok_dist = fabsf((float)(pred - gt));
        contrib = tok_dist * (sum_in / S) / (cnt1 + EPSF);
      } else {
        // mask1 term is 0; sum(mask2)=V-(re-rs);
        // sum_{mask2}(1+p) = cnt2 + (P_all - P_in).
        const float upper = *upper_p;
        const float cnt2 = (float)(VOCAB - (re - rs));
        const float pmass_out = (sum_all - sum_in) / S;
        contrib = upper * (cnt2 + pmass_out) / (cnt2 + EPSF);
      }
    }
  }

  if (lane == 0) wave_out[wv] = contrib;
  __syncthreads();
  if (threadIdx.x == 0) {
    float s = 0.0f;
#pragma unroll
    for (int i = 0; i < ROWS_PER_BLOCK; ++i) s += wave_out[i];
    partials[blockIdx.x] = s;  // deterministic per-block partial
  }
}

// --- Kernel C: deterministic final reduction of block partials --------------
__global__ void finalize_kernel(const float* __restrict__ partials, int n,
                                float* __restrict__ out) {
  __shared__ float sm[256];
  const int t = threadIdx.x;
  float s = 0.0f;
  for (int i = t; i < n; i += 256) s += partials[i];
  sm[t] = s;
  __syncthreads();
  for (int k = 128; k > 0; k >>= 1) {
    if (t < k) sm[t] += sm[t + k];
    __syncthreads();
  }
  if (t == 0) out[0] = sm[0];
}

extern "C" void kernel_launch(void* const* d_in, const int* in_sizes, int n_in,
                              void* d_out, int out_size, void* d_ws,
                              size_t ws_size, hipStream_t stream) {
  const float* logits = (const float*)d_in[0];
  const int* gts = (const int*)d_in[1];
  const int* event_ids = (const int*)d_in[2];
  const int* range_start = (const int*)d_in[3];
  const int* range_end = (const int*)d_in[4];

  const int n = in_sizes[1];  // N (ground_truths count)
  const int v = in_sizes[2];  // V (vocab) -- kernel compiled for V==1024

  float* upper = (float*)d_ws;                       // 1 float
  float* partials = (float*)((char*)d_ws + 256);     // nblocks floats

  const int nblocks = (n + ROWS_PER_BLOCK - 1) / ROWS_PER_BLOCK;

  upper_bound_kernel<<<1, 256, 0, stream>>>(range_start, range_end, upper, v);
  loss_main_kernel<<<nblocks, 256, 0, stream>>>(logits, gts, event_ids,
                                                range_start, range_end, upper,
                                                partials, n);
  finalize_kernel<<<1, 256, 0, stream>>>(partials, nblocks, (float*)d_out);
}